// WeightedRigidAlign_72318659330499
// MI455X (gfx1250) — compile-verified
//
#include <hip/hip_runtime.h>
#include <hip/hip_bf16.h>
#include <stdint.h>

typedef __attribute__((ext_vector_type(2))) float v2f;
typedef __attribute__((ext_vector_type(8))) float v8f;
typedef __attribute__((ext_vector_type(4))) unsigned int v4u;
typedef __attribute__((ext_vector_type(8))) int v8i;
typedef __attribute__((ext_vector_type(4))) int v4i;

#define WPB 8            // waves per block
#define SLICE 128        // points per wave slice
#define SUBT (SLICE/16)  // 16-point WMMA sub-tiles per slice

// Workspace float layout:
//  [0..15]  acc[i*4+j] = sum_n w * a_i * b_j   with a=(x0,x1,x2,1), b=(g0,g1,g2,1)
//  [16..24] R row-major
//  [25..27] t = mu - R*mu_gt

__global__ void wra_zero_ws(float* ws) {
    if (threadIdx.x < 16) ws[threadIdx.x] = 0.0f;
}

// TDM: 1-D copy of `tile_elems` f32 from global to LDS, expressed as a 2-D
// tile (tile_dim1 = 1). tensor_dim0 = remaining valid elements so that
// out-of-bounds reads return zero (free tail handling).
__device__ __forceinline__ void tdm_load_1d(unsigned lds_byte, const void* gptr,
                                            unsigned tile_elems, unsigned tensor_elems) {
    const unsigned long long ga = (unsigned long long)(uintptr_t)gptr;
    v4u g0;
    g0.x = 1u;                                                 // count=1, user mode
    g0.y = lds_byte;                                           // lds_addr
    g0.z = (unsigned)ga;                                       // global_addr[31:0]
    g0.w = (unsigned)((ga >> 32) & 0x01ffffffu) | (2u << 30);  // addr[56:32] | type=2
    v8i g1;
    g1[0] = (int)(2u << 16);                                   // data_size = 4B
    g1[1] = (int)((tensor_elems & 0xffffu) << 16);             // tensor_dim0[15:0]
    g1[2] = (int)((tensor_elems >> 16) & 0xffffu)              // tensor_dim0[31:16]
          | (int)(1u << 16);                                   // tensor_dim1 = 1
    g1[3] = (int)((tile_elems & 0xffffu) << 16);               // tile_dim0
    g1[4] = 1;                                                 // tile_dim1 = 1
    g1[5] = (int)tensor_elems;                                 // dim0_stride[31:0]
    g1[6] = 0;
    g1[7] = 0;
    const v4i gz4 = {0, 0, 0, 0};
    const v8i gz8 = {0, 0, 0, 0, 0, 0, 0, 0};
    __builtin_amdgcn_tensor_load_to_lds(g0, g1, gz4, gz4, gz8, 0);
}

// ---------------------------------------------------------------------------
// Pass 1: TDM double-buffered staging into per-wave LDS slices, consumed by
// v_wmma_f32_16x16x4_f32 accumulation (16 points/WMMA: 4 streams x K=4; the
// accumulator's 4 diagonal 4x4 blocks are valid partial Grams).
// ---------------------------------------------------------------------------
__global__ __launch_bounds__(256) void wra_reduce(const float* __restrict__ x,
                                                  const float* __restrict__ xgt,
                                                  const float* __restrict__ w,
                                                  float* __restrict__ ws, int N) {
    __shared__ float sx[WPB][2][3 * SLICE];
    __shared__ float sg[WPB][2][3 * SLICE];
    __shared__ float sw[WPB][2][SLICE];
    __shared__ float lacc[16];
    if (threadIdx.x < 16) lacc[threadIdx.x] = 0.0f;
    __syncthreads();

    const int lane = threadIdx.x & 31;
    const int half = lane >> 4;        // K pair {0,1} vs {2,3}
    const int q    = lane & 15;        // A row m / B col n
    const int s    = q >> 2;           // point-stream (0..3)
    const int comp = q & 3;            // component i == j
    const int c3   = comp < 3 ? comp : 2;
    const int wid  = threadIdx.x >> 5;

    const int gw = __builtin_amdgcn_readfirstlane(blockIdx.x * WPB + wid);
    const int nw = gridDim.x * WPB;

    v8f acc = {};

    int sl = gw;
    bool have = (sl * SLICE) < N;
    int buf = 0;
    if (have) {
        const int base = sl * SLICE;
        const unsigned rem = (unsigned)(N - base);
        tdm_load_1d((unsigned)(uintptr_t)&sx[wid][0][0], x   + 3 * (size_t)base, 3 * SLICE, rem * 3u);
        tdm_load_1d((unsigned)(uintptr_t)&sg[wid][0][0], xgt + 3 * (size_t)base, 3 * SLICE, rem * 3u);
        tdm_load_1d((unsigned)(uintptr_t)&sw[wid][0][0], w   + (size_t)base,     SLICE,     rem);
    }

    while (have) {
        const int nsl = sl + nw;
        const bool nhave = (nsl * SLICE) < N;
        if (nhave) {                       // prefetch next slice into other buffer
            const int nb   = buf ^ 1;
            const int base = nsl * SLICE;
            const unsigned rem = (unsigned)(N - base);
            tdm_load_1d((unsigned)(uintptr_t)&sx[wid][nb][0], x   + 3 * (size_t)base, 3 * SLICE, rem * 3u);
            tdm_load_1d((unsigned)(uintptr_t)&sg[wid][nb][0], xgt + 3 * (size_t)base, 3 * SLICE, rem * 3u);
            tdm_load_1d((unsigned)(uintptr_t)&sw[wid][nb][0], w   + (size_t)base,     SLICE,     rem);
            __builtin_amdgcn_s_wait_tensorcnt(3);   // current buffer's 3 done
        } else {
            __builtin_amdgcn_s_wait_tensorcnt(0);
        }

        const float* sxv = &sx[wid][buf][0];
        const float* sgv = &sg[wid][buf][0];
        const float* swv = &sw[wid][buf][0];
#pragma unroll
        for (int t = 0; t < SUBT; ++t) {
            const int pl0 = t * 16 + 4 * s + 2 * half;
            const int pl1 = pl0 + 1;
            const float w0  = swv[pl0];
            const float w1  = swv[pl1];
            const float xa0 = sxv[3 * pl0 + c3];
            const float xa1 = sxv[3 * pl1 + c3];
            const float ga0 = sgv[3 * pl0 + c3];
            const float ga1 = sgv[3 * pl1 + c3];
            v2f a, b;                                   // branchless fragments
            a.x = comp < 3 ? w0 * xa0 : w0;
            a.y = comp < 3 ? w1 * xa1 : w1;
            b.x = comp < 3 ? ga0 : 1.0f;
            b.y = comp < 3 ? ga1 : 1.0f;
            acc = __builtin_amdgcn_wmma_f32_16x16x4_f32(false, a, false, b,
                                                        (short)0, acc, false, false);
        }
        sl = nsl; have = nhave; buf ^= 1;
    }

    // Fold diagonal stream blocks: acc[v] = D[m][q], m = v + 8*half.
#pragma unroll
    for (int v = 0; v < 8; ++v) {
        const int m = v + 8 * half;
        if ((m >> 2) == s) atomicAdd(&lacc[(m & 3) * 4 + comp], acc[v]);
    }
    __syncthreads();
    if (threadIdx.x < 16) atomicAdd(&ws[threadIdx.x], lacc[threadIdx.x]);
}

// ---------------------------------------------------------------------------
// Pass 2: 3x3 Kabsch solve on one lane. Jacobi eigen of K = H^T H -> V,
// U via normalize / Gram-Schmidt / cross so det-correction = det(V).
// ---------------------------------------------------------------------------
__global__ void wra_solve(float* ws) {
    if (threadIdx.x != 0) return;

    float acc[16];
#pragma unroll
    for (int i = 0; i < 16; ++i) acc[i] = ws[i];

    const float Sw = acc[15];
    const float inv_ws = 1.0f / (Sw + 1e-8f);
    const float Swx[3] = { acc[3],  acc[7],  acc[11] };
    const float Swg[3] = { acc[12], acc[13], acc[14] };
    float mu[3], mug[3];
#pragma unroll
    for (int i = 0; i < 3; ++i) { mu[i] = Swx[i] * inv_ws; mug[i] = Swg[i] * inv_ws; }

    float H[3][3];
#pragma unroll
    for (int i = 0; i < 3; ++i)
#pragma unroll
        for (int j = 0; j < 3; ++j)
            H[i][j] = acc[i * 4 + j] - Swx[i] * mug[j] - mu[i] * Swg[j] + Sw * mu[i] * mug[j];

    float K[3][3];
#pragma unroll
    for (int a = 0; a < 3; ++a)
#pragma unroll
        for (int b = 0; b < 3; ++b) {
            float t = 0.0f;
#pragma unroll
            for (int r = 0; r < 3; ++r) t += H[r][a] * H[r][b];
            K[a][b] = t;
        }

    float V[3][3] = { {1,0,0}, {0,1,0}, {0,0,1} };
    for (int sweep = 0; sweep < 15; ++sweep) {
        for (int pi = 0; pi < 3; ++pi) {
            const int p = (pi == 2) ? 1 : 0;
            const int q = (pi == 0) ? 1 : 2;
            const float apq = K[p][q];
            if (fabsf(apq) < 1e-20f) continue;
            const float theta = (K[q][q] - K[p][p]) / (2.0f * apq);
            const float tt  = copysignf(1.0f, theta) / (fabsf(theta) + sqrtf(1.0f + theta * theta));
            const float cth = 1.0f / sqrtf(1.0f + tt * tt);
            const float sth = tt * cth;
            for (int k = 0; k < 3; ++k) {
                const float kp = K[p][k], kq = K[q][k];
                K[p][k] = cth * kp - sth * kq;
                K[q][k] = sth * kp + cth * kq;
            }
            for (int k = 0; k < 3; ++k) {
                const float kp = K[k][p], kq = K[k][q];
                K[k][p] = cth * kp - sth * kq;
                K[k][q] = sth * kp + cth * kq;
            }
            for (int k = 0; k < 3; ++k) {
                const float vp = V[k][p], vq = V[k][q];
                V[k][p] = cth * vp - sth * vq;
                V[k][q] = sth * vp + cth * vq;
            }
        }
    }

    float lam[3] = { K[0][0], K[1][1], K[2][2] };
    int idx[3] = { 0, 1, 2 };
    for (int i = 0; i < 2; ++i)
        for (int j = i + 1; j < 3; ++j)
            if (lam[idx[j]] > lam[idx[i]]) { int t = idx[i]; idx[i] = idx[j]; idx[j] = t; }

    float v0[3], v1[3], v2[3];
#pragma unroll
    for (int r = 0; r < 3; ++r) { v0[r] = V[r][idx[0]]; v1[r] = V[r][idx[1]]; v2[r] = V[r][idx[2]]; }

    float u0[3], u1[3], u2[3];
#pragma unroll
    for (int r = 0; r < 3; ++r) u0[r] = H[r][0]*v0[0] + H[r][1]*v0[1] + H[r][2]*v0[2];
    const float n0 = sqrtf(u0[0]*u0[0] + u0[1]*u0[1] + u0[2]*u0[2]) + 1e-20f;
#pragma unroll
    for (int r = 0; r < 3; ++r) u0[r] /= n0;
#pragma unroll
    for (int r = 0; r < 3; ++r) u1[r] = H[r][0]*v1[0] + H[r][1]*v1[1] + H[r][2]*v1[2];
    const float d01 = u0[0]*u1[0] + u0[1]*u1[1] + u0[2]*u1[2];
#pragma unroll
    for (int r = 0; r < 3; ++r) u1[r] -= d01 * u0[r];
    const float n1 = sqrtf(u1[0]*u1[0] + u1[1]*u1[1] + u1[2]*u1[2]) + 1e-20f;
#pragma unroll
    for (int r = 0; r < 3; ++r) u1[r] /= n1;
    u2[0] = u0[1]*u1[2] - u0[2]*u1[1];
    u2[1] = u0[2]*u1[0] - u0[0]*u1[2];
    u2[2] = u0[0]*u1[1] - u0[1]*u1[0];

    const float detV = v0[0]*(v1[1]*v2[2] - v1[2]*v2[1])
                     - v0[1]*(v1[0]*v2[2] - v1[2]*v2[0])
                     + v0[2]*(v1[0]*v2[1] - v1[1]*v2[0]);

    float R[3][3];
#pragma unroll
    for (int i = 0; i < 3; ++i)
#pragma unroll
        for (int j = 0; j < 3; ++j)
            R[i][j] = u0[i]*v0[j] + u1[i]*v1[j] + detV * u2[i]*v2[j];

#pragma unroll
    for (int i = 0; i < 3; ++i) {
#pragma unroll
        for (int j = 0; j < 3; ++j) ws[16 + 3*i + j] = R[i][j];
        ws[25 + i] = mu[i] - (R[i][0]*mug[0] + R[i][1]*mug[1] + R[i][2]*mug[2]);
    }
}

// ---------------------------------------------------------------------------
// Pass 3: out = x_gt * R^T + t as a WMMA matmul. A = [R | t] rows (constant
// fragment), B cols = (x_gt,1) of 16 points, D rows 0..2 = output xyz.
// ---------------------------------------------------------------------------
__global__ __launch_bounds__(256) void wra_apply(const float* __restrict__ xgt,
                                                 const float* __restrict__ ws,
                                                 float* __restrict__ out, int N) {
    const int lane = threadIdx.x & 31;
    const int half = lane >> 4;
    const int m    = lane & 15;

    v2f a = {};
    if (m < 3) {
        if (half == 0) { a.x = ws[16 + 3*m + 0]; a.y = ws[16 + 3*m + 1]; }
        else           { a.x = ws[16 + 3*m + 2]; a.y = ws[25 + m];       }
    }

    const int gw = __builtin_amdgcn_readfirstlane(blockIdx.x * WPB + (threadIdx.x >> 5));
    const int nw = gridDim.x * WPB;
    const int stride = nw * 16;

    for (int base = gw * 16; base < N; base += stride) {   // wave-uniform bound
        const int p  = base + m;
        const int pc = p < N ? p : N - 1;
        __builtin_prefetch(xgt + 3 * pc + 3 * stride, 0, 0);

        const float e0 = xgt[3 * pc + 2 * half];   // comp0 (half0) or comp2 (half1)
        const float e1 = xgt[3 * pc + 1];
        v2f b;
        b.x = e0;
        b.y = half ? 1.0f : e1;

        v8f c = {};
        c = __builtin_amdgcn_wmma_f32_16x16x4_f32(false, a, false, b,
                                                  (short)0, c, false, false);
        if (half == 0 && p < N) {
            __builtin_nontemporal_store(c[0], out + 3 * p + 0);
            __builtin_nontemporal_store(c[1], out + 3 * p + 1);
            __builtin_nontemporal_store(c[2], out + 3 * p + 2);
        }
    }
}

extern "C" void kernel_launch(void* const* d_in, const int* in_sizes, int n_in,
                              void* d_out, int out_size, void* d_ws, size_t ws_size,
                              hipStream_t stream) {
    const float* x   = (const float*)d_in[0];
    const float* xgt = (const float*)d_in[1];
    const float* w   = (const float*)d_in[2];
    float* out = (float*)d_out;
    float* ws  = (float*)d_ws;
    const int N = in_sizes[2];   // weights count == number of points

    wra_zero_ws<<<1, 32, 0, stream>>>(ws);
    wra_reduce<<<512, 256, 0, stream>>>(x, xgt, w, ws, N);
    wra_solve<<<1, 32, 0, stream>>>(ws);
    wra_apply<<<2048, 256, 0, stream>>>(xgt, ws, out, N);
}